// HGCF_39238821216529
// MI455X (gfx1250) — compile-verified
//
#include <hip/hip_runtime.h>
#include <math.h>

// ---------------------------------------------------------------------------
// Hyperbolic GCN pipeline for MI455X (gfx1250, wave32).
//   A = logmap0(proj(x));  s1 = A*adj; s2 = s1*adj; s3 = s2*adj (scatter spmm)
//   out = proj(expmap0(s1+s2+s3))
// SPMM uses the CDNA5 async global->LDS engine (ASYNCcnt) to stage gathered
// rows, then scatters with agent-scope f32 atomic adds. Edge metadata is
// broadcast with v_readlane (wave-uniform) instead of ds_bpermute.
// ---------------------------------------------------------------------------

#define GLOBAL_AS __attribute__((address_space(1)))
#define LDS_AS    __attribute__((address_space(3)))

typedef int v2i __attribute__((vector_size(8)));  // builtin's pointee type

#ifndef D
#define D 64
#endif

__device__ __forceinline__ float waveReduceSum(float v) {
  v += __shfl_xor(v, 16, 32);
  v += __shfl_xor(v, 8, 32);
  v += __shfl_xor(v, 4, 32);
  v += __shfl_xor(v, 2, 32);
  v += __shfl_xor(v, 1, 32);
  return v;
}

__device__ __forceinline__ int rl_i(int v, int l) {
  return __builtin_amdgcn_readlane(v, l);
}
__device__ __forceinline__ float rl_f(float v, int l) {
  return __int_as_float(__builtin_amdgcn_readlane(__float_as_int(v), l));
}

__device__ __forceinline__ void atomic_add_f32(float* p, float v) {
  __hip_atomic_fetch_add(p, v, __ATOMIC_RELAXED, __HIP_MEMORY_SCOPE_AGENT);
}

__device__ __forceinline__ void async_copy_row_b64(const float* gsrc, float* ldst) {
  // per-lane 8-byte async copy, 32 lanes => one contiguous 256B row
  __builtin_amdgcn_global_load_async_to_lds_b64(
      (GLOBAL_AS v2i*)gsrc, (LDS_AS v2i*)ldst, 0, 0);
}

// ---- kernel 1: A = logmap0(proj(x)), one wave per row --------------------
__global__ __launch_bounds__(256) void k_xt(const float* __restrict__ x,
                                            float* __restrict__ out, int n) {
  int w = (blockIdx.x * blockDim.x + threadIdx.x) >> 5;
  int lane = threadIdx.x & 31;
  if (w >= n) return;
  const float2 v = *(const float2*)(x + (size_t)w * D + 2 * lane);
  float a = (lane == 0) ? 0.0f : v.x;          // dim 0 excluded from y
  float sq = waveReduceSum(a * a + v.y * v.y); // ||y||^2
  float x0 = sqrtf(fmaxf(1.0f + sq, 1e-7f));   // proj component 0 (K=1)
  float ynorm = fmaxf(sqrtf(sq), 1e-15f);
  float theta = fmaxf(x0, 1.0f + 1e-7f);
  float s = acoshf(theta) / ynorm;
  float2 o;
  o.x = a * s;   // lane0 -> 0 (logmap0 zeros dim 0)
  o.y = v.y * s;
  *(float2*)(out + (size_t)w * D + 2 * lane) = o;
}

// ---- kernel 2: scatter SPMM, 32 edges per wave, async-staged gathers -----
__global__ __launch_bounds__(256) void k_spmm(const int* __restrict__ rows,
                                              const int* __restrict__ cols,
                                              const float* __restrict__ vals,
                                              const float* __restrict__ X,
                                              float* __restrict__ Y, int E) {
  // 8 waves * 2 buffers * 16 edges * 64 floats * 4B = 64 KB
  __shared__ float smem[8 * 2 * 16 * D];
  const int lane = threadIdx.x & 31;
  const int wave = threadIdx.x >> 5;
  const int gwave = blockIdx.x * 8 + wave;
  const int e0 = gwave * 32;
  if (e0 >= E) return;
  const int cnt = (E - e0 < 32) ? (E - e0) : 32;

  int myRow = 0, myCol = 0;
  float myVal = 0.0f;
  if (lane < cnt) {
    myRow = rows[e0 + lane];
    myCol = cols[e0 + lane];
    myVal = vals[e0 + lane];
  }

  float* wb = smem + wave * (2 * 16 * D); // this wave's double buffer

  if (cnt == 32) {
    // stage buffer 0 (edges 0..15) then buffer 1 (edges 16..31)
    #pragma unroll
    for (int j = 0; j < 16; ++j) {
      int c = rl_i(myCol, j);                    // wave-uniform column
      async_copy_row_b64(X + (size_t)c * D + 2 * lane, wb + j * D + 2 * lane);
    }
    #pragma unroll
    for (int j = 16; j < 32; ++j) {
      int c = rl_i(myCol, j);
      async_copy_row_b64(X + (size_t)c * D + 2 * lane,
                         wb + 16 * D + (j - 16) * D + 2 * lane);
    }
    __builtin_amdgcn_s_wait_asynccnt(16); // buffer 0 complete (in-order)
    asm volatile("" ::: "memory");
    #pragma unroll
    for (int j = 0; j < 16; ++j) {
      int r = rl_i(myRow, j);                    // wave-uniform row -> saddr
      float vv = rl_f(myVal, j);
      float2 xv = *(float2*)(wb + j * D + 2 * lane);
      float* yp = Y + (size_t)r * D + 2 * lane;
      atomic_add_f32(yp, vv * xv.x);
      atomic_add_f32(yp + 1, vv * xv.y);
    }
    __builtin_amdgcn_s_wait_asynccnt(0); // buffer 1 complete
    asm volatile("" ::: "memory");
    #pragma unroll
    for (int j = 16; j < 32; ++j) {
      int r = rl_i(myRow, j);
      float vv = rl_f(myVal, j);
      float2 xv = *(float2*)(wb + 16 * D + (j - 16) * D + 2 * lane);
      float* yp = Y + (size_t)r * D + 2 * lane;
      atomic_add_f32(yp, vv * xv.x);
      atomic_add_f32(yp + 1, vv * xv.y);
    }
  } else {
    // tail batch: plain gathers (E divisible by 32 for this problem anyway)
    for (int j = 0; j < cnt; ++j) {
      int r = rl_i(myRow, j);
      int c = rl_i(myCol, j);
      float vv = rl_f(myVal, j);
      float2 xv = *(const float2*)(X + (size_t)c * D + 2 * lane);
      float* yp = Y + (size_t)r * D + 2 * lane;
      atomic_add_f32(yp, vv * xv.x);
      atomic_add_f32(yp + 1, vv * xv.y);
    }
  }
}

// ---- small elementwise helpers -------------------------------------------
__global__ __launch_bounds__(256) void k_zero4(float4* __restrict__ p, int n4) {
  for (int i = blockIdx.x * blockDim.x + threadIdx.x; i < n4;
       i += gridDim.x * blockDim.x) {
    p[i] = make_float4(0.f, 0.f, 0.f, 0.f);
  }
}

__global__ __launch_bounds__(256) void k_acc4(const float4* __restrict__ s,
                                              float4* __restrict__ d, int n4) {
  for (int i = blockIdx.x * blockDim.x + threadIdx.x; i < n4;
       i += gridDim.x * blockDim.x) {
    float4 a = d[i];
    float4 b = s[i];
    a.x += b.x; a.y += b.y; a.z += b.z; a.w += b.w;
    d[i] = a;
  }
}

// ---- kernel 3: out = proj(expmap0(h)) in place, one wave per row ---------
__global__ __launch_bounds__(256) void k_final(float* __restrict__ h, int n) {
  int w = (blockIdx.x * blockDim.x + threadIdx.x) >> 5;
  int lane = threadIdx.x & 31;
  if (w >= n) return;
  float* rowp = h + (size_t)w * D + 2 * lane;
  const float2 v = *(const float2*)rowp;
  float a = (lane == 0) ? 0.0f : v.x;            // expmap0 uses dims 1..63
  float sq = waveReduceSum(a * a + v.y * v.y);
  float xnorm = fmaxf(sqrtf(sq), 1e-15f);
  float theta = xnorm;                            // sqrtK = 1
  float s = sinhf(theta) / xnorm;
  float r1x = a * s;
  float r1y = v.y * s;
  // proj of [cosh, res1]: component 0 recomputed from ||res1||^2
  float sq2 = waveReduceSum(r1x * r1x + r1y * r1y);
  float x0 = sqrtf(fmaxf(1.0f + sq2, 1e-7f));
  float2 o;
  o.x = (lane == 0) ? x0 : r1x;
  o.y = r1y;
  *(float2*)rowp = o;
}

// ---------------------------------------------------------------------------
extern "C" void kernel_launch(void* const* d_in, const int* in_sizes, int n_in,
                              void* d_out, int out_size, void* d_ws, size_t ws_size,
                              hipStream_t stream) {
  const float* x    = (const float*)d_in[0];
  const int*   rows = (const int*)d_in[1];
  const int*   cols = (const int*)d_in[2];
  const float* vals = (const float*)d_in[3];
  float* out = (float*)d_out;

  const int nelem = in_sizes[0];     // N * 64
  const int n = nelem / D;           // N rows
  const int E = in_sizes[1];

  float* A = (float*)d_ws;           // N*64 floats
  float* B = A + (size_t)nelem;      // N*64 floats

  const int rowBlocks  = (n * 32 + 255) / 256;              // wave per row
  const int n4         = nelem / 4;
  const int elemBlocks = (n4 + 255) / 256;
  const int waves      = (E + 31) / 32;
  const int spmmBlocks = (waves + 7) / 8;

  // A = logmap0(proj(x))
  k_xt<<<rowBlocks, 256, 0, stream>>>(x, A, n);

  // s1 -> d_out
  k_zero4<<<elemBlocks, 256, 0, stream>>>((float4*)out, n4);
  k_spmm<<<spmmBlocks, 256, 0, stream>>>(rows, cols, vals, A, out, E);

  // s2 -> B
  k_zero4<<<elemBlocks, 256, 0, stream>>>((float4*)B, n4);
  k_spmm<<<spmmBlocks, 256, 0, stream>>>(rows, cols, vals, out, B, E);

  // s3 -> A (A no longer needed as input)
  k_zero4<<<elemBlocks, 256, 0, stream>>>((float4*)A, n4);
  k_spmm<<<spmmBlocks, 256, 0, stream>>>(rows, cols, vals, B, A, E);

  // h = s1 + s2 + s3 accumulated into d_out
  k_acc4<<<elemBlocks, 256, 0, stream>>>((const float4*)B, (float4*)out, n4);
  k_acc4<<<elemBlocks, 256, 0, stream>>>((const float4*)A, (float4*)out, n4);

  // out = proj(expmap0(h)) in place
  k_final<<<rowBlocks, 256, 0, stream>>>(out, n);
}